// FlowGNN_28647431865082
// MI455X (gfx1250) — compile-verified
//
#include <hip/hip_runtime.h>
#include <stdint.h>

#define NUM_LAYER 6
#define NUM_PATHS 4
#define NUM_PATH_NODE 400000
#define NUM_EDGE_NODE 20000
#define NTOT (NUM_EDGE_NODE + NUM_PATH_NODE)   // 420000
#define NEDGE 12000000
#define NROWS_DNN (NUM_PATH_NODE / NUM_PATHS)  // 100000
#define MAXW 24

typedef __attribute__((ext_vector_type(2))) float v2f;
typedef __attribute__((ext_vector_type(8))) float v8f;

#if __has_builtin(__builtin_amdgcn_global_load_async_to_lds_b32)
#define HAVE_ASYNC_G2L 1
#endif

// Stage one dword global -> LDS. Async (ASYNCcnt) when the gfx1250 builtin
// exists; plain copy fallback otherwise.
__device__ __forceinline__ void stage_g2l(const float* gsrc, float* ldst) {
#ifdef HAVE_ASYNC_G2L
    __builtin_amdgcn_global_load_async_to_lds_b32(
        (__attribute__((address_space(1))) int*)(uintptr_t)gsrc,
        (__attribute__((address_space(3))) int*)(uint32_t)(uintptr_t)ldst,
        0, 0);
#else
    *ldst = *gsrc;
#endif
}

__device__ __forceinline__ void wait_async0() {
#if __has_builtin(__builtin_amdgcn_s_wait_asynccnt)
    __builtin_amdgcn_s_wait_asynccnt(0);
#else
    asm volatile("s_wait_asynccnt 0x0" ::: "memory");
#endif
}

// ---------------- per-node linear: h2 = h @ w.T + b ----------------
template<int D>
__global__ void node_linear_kernel(const float* __restrict__ h,
                                   const float* __restrict__ w,
                                   const float* __restrict__ b,
                                   float* __restrict__ h2) {
    int n = blockIdx.x * blockDim.x + threadIdx.x;
    if (n >= NTOT) return;
    float hv[D];
#pragma unroll
    for (int k = 0; k < D; ++k) hv[k] = h[(size_t)n * D + k];
#pragma unroll
    for (int j = 0; j < D; ++j) {
        float s = b[j];
#pragma unroll
        for (int k = 0; k < D; ++k) s += hv[k] * w[j * D + k];
        h2[(size_t)n * D + j] = s;
    }
}

__global__ void zero_kernel(float* __restrict__ p, int n) {
    int i = blockIdx.x * blockDim.x + threadIdx.x;
    if (i < n) p[i] = 0.0f;
}

// ---------------- edge scatter: acc[row] += h2[col] * ev ----------------
template<int D>
__global__ void edge_kernel(const float* __restrict__ h2,
                            const int* __restrict__ row,
                            const int* __restrict__ col,
                            const float* __restrict__ ev,
                            float* __restrict__ acc) {
    int e = blockIdx.x * blockDim.x + threadIdx.x;
    if (e >= NEDGE) return;
    // one speculative prefetch per 64 edges, 64KB ahead (global_prefetch_b8)
    if ((e & 63) == 0) {
        __builtin_prefetch(row + e + 16384, 0, 1);
        __builtin_prefetch(col + e + 16384, 0, 1);
        __builtin_prefetch(ev + e + 16384, 0, 1);
    }
    int r = row[e];
    int c = col[e];
    float v = ev[e];
#pragma unroll
    for (int k = 0; k < D; ++k)
        atomicAdd(&acc[(size_t)r * D + k], h2[(size_t)c * D + k] * v);
}

// ---------------- DNN GEMM on path nodes via V_WMMA_F32_16X16X4_F32 ----------------
// hp: (NROWS_DNN, W) flat (== acc path region, contiguous since W = 4*D)
// out[r, j] = sum_k hp[r,k] * wd[j,k] + bd[j], scattered into hnext (stride D+1).
// wd/bd staged into LDS once per workgroup via async global->LDS.
__global__ void dnn_wmma_kernel(const float* __restrict__ hp,
                                const float* __restrict__ wd,
                                const float* __restrict__ bd,
                                float* __restrict__ hnext,
                                int W, int D, int tilesN) {
    __shared__ float smem[MAXW * MAXW + MAXW];   // wd (W*W) then bd (W)
    const int tid = threadIdx.x;
    const int WW = W * W;
    const int total = WW + W;
    for (int idx = tid; idx < total; idx += 128) {
        const float* src = (idx < WW) ? (wd + idx) : (bd + (idx - WW));
        stage_g2l(src, &smem[idx]);
    }
#ifdef HAVE_ASYNC_G2L
    wait_async0();
#endif
    __syncthreads();

    int wid = tid >> 5;
    int lane = tid & 31;
    int gw = blockIdx.x * 4 + wid;
    int tiles = (NROWS_DNN / 16) * tilesN;
    if (gw >= tiles) return;              // wave-uniform: EXEC all-ones at WMMA
    int mtile = gw / tilesN;
    int ntile = gw - mtile * tilesN;
    int half = lane >> 4;                 // 0: lanes 0-15, 1: lanes 16-31
    int l16 = lane & 15;
    int row0 = mtile * 16;
    int j = ntile * 16 + l16;             // output column (N index)
    bool jv = (j < W);
    int js = jv ? j : (W - 1);            // clamp: garbage columns are never stored

    v8f c = {};
    for (int kc = 0; kc < W; kc += 4) {
        int ka = kc + 2 * half;           // A/B VGPR0 holds K=0 (lo half) / K=2 (hi half)
        v2f a, b;
        const float* ap = hp + (size_t)(row0 + l16) * W + ka;
        a.x = ap[0];
        a.y = ap[1];
        const float* bp = smem + js * W + ka;    // B[k][j] = wd[j*W + k] (from LDS)
        b.x = bp[0];
        b.y = bp[1];
        // 8 args: (neg_a, A, neg_b, B, c_mod, C, reuse_a, reuse_b)
        c = __builtin_amdgcn_wmma_f32_16x16x4_f32(false, a, false, b,
                                                  (short)0, c, false, false);
    }
    if (jv) {
        float bias = smem[WW + j];
        int kdiv = j / D;                 // which of the 4 grouped path nodes
        int kmod = j - kdiv * D;          // feature within node
#pragma unroll
        for (int v = 0; v < 8; ++v) {
            int r = row0 + v + 8 * half;  // C/D: VGPR v holds M=v (lo) / M=v+8 (hi)
            int node = NUM_EDGE_NODE + 4 * r + kdiv;
            hnext[(size_t)node * (D + 1) + kmod] = c[v] + bias;
        }
    }
}

// ---------------- finalize: edge-node features + h_0 concat column ----------------
__global__ void finalize_kernel(const float* __restrict__ acc,
                                const float* __restrict__ h0,
                                float* __restrict__ hnext, int D) {
    int n = blockIdx.x * blockDim.x + threadIdx.x;
    if (n >= NTOT) return;
    if (n < NUM_EDGE_NODE) {
        for (int cc = 0; cc < D; ++cc)
            hnext[(size_t)n * (D + 1) + cc] = acc[(size_t)n * D + cc];
    }
    hnext[(size_t)n * (D + 1) + D] = h0[n];
}

__global__ void copy_kernel(const float* __restrict__ src, float* __restrict__ dst, int n) {
    int i = blockIdx.x * blockDim.x + threadIdx.x;
    if (i < n) dst[i] = src[i];
}

extern "C" void kernel_launch(void* const* d_in, const int* in_sizes, int n_in,
                              void* d_out, int out_size, void* d_ws, size_t ws_size,
                              hipStream_t stream) {
    (void)in_sizes; (void)n_in; (void)out_size; (void)ws_size;
    const float* h0 = (const float*)d_in[0];
    const int* eidx = (const int*)d_in[1];
    const int* row = eidx;
    const int* col = eidx + NEDGE;
    const float* ev = (const float*)d_in[2];
    // flattened params: gnn (w_i,b_i) at d_in[3+2i]/d_in[4+2i]; dnn at d_in[15+2i]/d_in[16+2i]

    float* ws = (float*)d_ws;
    float* hA  = ws;                               // NTOT * 7
    float* hB  = hA + (size_t)NTOT * 7;            // NTOT * 7
    float* h2  = hB + (size_t)NTOT * 7;            // NTOT * 6
    float* acc = h2 + (size_t)NTOT * 6;            // NTOT * 6

    const float* hcur = h0;                        // layer 0 input, dim 1
    float* hnext = hA;

    const int nb = (NTOT + 255) / 256;
    const int eb = (NEDGE + 255) / 256;

    for (int i = 0; i < NUM_LAYER; ++i) {
        const int d = i + 1;
        const float* w  = (const float*)d_in[3 + 2 * i];
        const float* b  = (const float*)d_in[4 + 2 * i];
        const float* wd = (const float*)d_in[15 + 2 * i];
        const float* bd = (const float*)d_in[16 + 2 * i];

        switch (d) {
            case 1: node_linear_kernel<1><<<nb, 256, 0, stream>>>(hcur, w, b, h2); break;
            case 2: node_linear_kernel<2><<<nb, 256, 0, stream>>>(hcur, w, b, h2); break;
            case 3: node_linear_kernel<3><<<nb, 256, 0, stream>>>(hcur, w, b, h2); break;
            case 4: node_linear_kernel<4><<<nb, 256, 0, stream>>>(hcur, w, b, h2); break;
            case 5: node_linear_kernel<5><<<nb, 256, 0, stream>>>(hcur, w, b, h2); break;
            default: node_linear_kernel<6><<<nb, 256, 0, stream>>>(hcur, w, b, h2); break;
        }

        int nz = NTOT * d;
        zero_kernel<<<(nz + 255) / 256, 256, 0, stream>>>(acc, nz);

        switch (d) {
            case 1: edge_kernel<1><<<eb, 256, 0, stream>>>(h2, row, col, ev, acc); break;
            case 2: edge_kernel<2><<<eb, 256, 0, stream>>>(h2, row, col, ev, acc); break;
            case 3: edge_kernel<3><<<eb, 256, 0, stream>>>(h2, row, col, ev, acc); break;
            case 4: edge_kernel<4><<<eb, 256, 0, stream>>>(h2, row, col, ev, acc); break;
            case 5: edge_kernel<5><<<eb, 256, 0, stream>>>(h2, row, col, ev, acc); break;
            default: edge_kernel<6><<<eb, 256, 0, stream>>>(h2, row, col, ev, acc); break;
        }

        int W = 4 * d;
        int tilesN = (W + 15) / 16;                       // 1 for d<=4, 2 for d=5,6
        int tiles = (NROWS_DNN / 16) * tilesN;            // 100000/16 = 6250 exact
        int wblocks = (tiles + 3) / 4;                    // 4 waves per 128-thread block
        dnn_wmma_kernel<<<wblocks, 128, 0, stream>>>(
            acc + (size_t)NUM_EDGE_NODE * d, wd, bd, hnext, W, d, tilesN);

        finalize_kernel<<<nb, 256, 0, stream>>>(acc, h0, hnext, d);

        hcur = hnext;
        hnext = (hnext == hA) ? hB : hA;
    }

    // hcur now has dim 7; emit the last NUM_PATH_NODE rows (contiguous)
    int total = NUM_PATH_NODE * 7;
    copy_kernel<<<(total + 255) / 256, 256, 0, stream>>>(
        hcur + (size_t)NUM_EDGE_NODE * 7, (float*)d_out, total);
}